// QuadModalCodebook_10204842295882
// MI455X (gfx1250) — compile-verified
//
#include <hip/hip_runtime.h>
#include <math.h>

typedef float v2f __attribute__((ext_vector_type(2)));
typedef float v8f __attribute__((ext_vector_type(8)));

#define LN_EPS 1e-5f

// ---------------------------------------------------------------------------
// GEMM: C[M,N] = A[M,K] @ B[K,N] + bias[N]   (fp32, V_WMMA_F32_16X16X4_F32)
// grid = (N/64, M/64), block = 256 (8 waves; 4 row-tiles x 2 wave-cols).
// Each wave computes a 16x32 C tile (two 16x16 accumulators sharing the A
// fragment).  All addressing is pointer-recurrence so the k-loop strength-
// reduces to adds on loop-carried address registers.
// ---------------------------------------------------------------------------
__global__ __launch_bounds__(256) void gemm_f32_wmma(
    const float* __restrict__ A, const float* __restrict__ B,
    const float* __restrict__ bias, float* __restrict__ C,
    int K, int N)
{
  const int lane = threadIdx.x & 31;
  const int wave = threadIdx.x >> 5;
  const int row0 = blockIdx.y * 64 + (wave & 3) * 16;
  const int col0 = blockIdx.x * 64 + (wave >> 2) * 32;
  const int rsel = lane & 15;      // row (A) / col (B,C) selector within tile
  const int hsel = lane >> 4;      // half-wave: K offset +2 for A/B fragments

  const float* aptr = A + (size_t)(row0 + rsel) * K + 2 * hsel;
  const float* bptr = B + (size_t)(2 * hsel) * N + col0 + rsel;
  const size_t bstep = (size_t)4 * N;

  v8f acc0 = {}, acc1 = {};
#pragma unroll 4
  for (int k = 0; k < K; k += 4) {
    v2f a = *(const v2f*)aptr;                  // A[m][ka], A[m][ka+1]
    v2f b0, b1;
    b0.x = bptr[0];                             // B[ka][n]
    b0.y = bptr[N];                             // B[ka+1][n]
    b1.x = bptr[16];
    b1.y = bptr[N + 16];
    acc0 = __builtin_amdgcn_wmma_f32_16x16x4_f32(
        false, a, false, b0, (short)0, acc0, false, false);
    acc1 = __builtin_amdgcn_wmma_f32_16x16x4_f32(
        false, a, false, b1, (short)0, acc1, false, false);
    aptr += 4;
    bptr += bstep;
  }

  const float bv0 = bias ? bias[col0 + rsel] : 0.0f;
  const float bv1 = bias ? bias[col0 + 16 + rsel] : 0.0f;
#pragma unroll
  for (int v = 0; v < 8; ++v) {
    const size_t r = (size_t)(row0 + v + 8 * hsel) * N + col0 + rsel;
    C[r]      = acc0[v] + bv0;
    C[r + 16] = acc1[v] + bv1;
  }
}

// ---------------------------------------------------------------------------
// Row-wise LayerNorm (+ optional ReLU).  One block (256 thr) per row.
// W = 256 or 512.  In-place safe (X may equal Y).
// ---------------------------------------------------------------------------
__global__ __launch_bounds__(256) void layernorm_rows(
    const float* __restrict__ X, const float* __restrict__ g,
    const float* __restrict__ be, float* __restrict__ Y,
    int W, int relu)
{
  __shared__ float red[8];
  const int tid  = threadIdx.x;
  const int lane = tid & 31, w = tid >> 5;
  const size_t row = blockIdx.x;
  const float* x = X + row * W;

  const float v0 = x[tid];
  const float v1 = (W > 256) ? x[tid + 256] : 0.0f;

  float s = v0 + v1;
  for (int off = 16; off; off >>= 1) s += __shfl_xor(s, off, 32);
  if (!lane) red[w] = s;
  __syncthreads();
  float mu = 0.0f;
#pragma unroll
  for (int i = 0; i < 8; ++i) mu += red[i];
  mu *= (1.0f / (float)W);
  __syncthreads();

  const float d0 = v0 - mu;
  const float d1 = (W > 256) ? (v1 - mu) : 0.0f;
  float s2 = d0 * d0 + d1 * d1;
  for (int off = 16; off; off >>= 1) s2 += __shfl_xor(s2, off, 32);
  if (!lane) red[w] = s2;
  __syncthreads();
  float var = 0.0f;
#pragma unroll
  for (int i = 0; i < 8; ++i) var += red[i];
  var *= (1.0f / (float)W);

  const float inv = rsqrtf(var + LN_EPS);
  float o0 = d0 * inv * g[tid] + be[tid];
  if (relu) o0 = fmaxf(o0, 0.0f);
  Y[row * W + tid] = o0;
  if (W > 256) {
    float o1 = d1 * inv * g[tid + 256] + be[tid + 256];
    if (relu) o1 = fmaxf(o1, 0.0f);
    Y[row * W + tid + 256] = o1;
  }
}

// ---------------------------------------------------------------------------
// ce[k] = ||E_k||^2.  grid = K, block = 256 (CD == 256).
// ---------------------------------------------------------------------------
__global__ __launch_bounds__(256) void codebook_norms(
    const float* __restrict__ E, float* __restrict__ ce)
{
  __shared__ float red[8];
  const int tid = threadIdx.x, lane = tid & 31, w = tid >> 5;
  const float v = E[(size_t)blockIdx.x * 256 + tid];
  float s = v * v;
  for (int off = 16; off; off >>= 1) s += __shfl_xor(s, off, 32);
  if (!lane) red[w] = s;
  __syncthreads();
  if (tid == 0) {
    float t = 0.0f;
#pragma unroll
    for (int i = 0; i < 8; ++i) t += red[i];
    ce[blockIdx.x] = t;
  }
}

// ---------------------------------------------------------------------------
// VQ argmin:  dist[n,k] = ||e_k||^2 - 2 z_n.e_k  (||z||^2 constant per row).
// Block owns 16 rows of z staged in padded LDS.  8 waves split 256 pairs of
// 16-code tiles; each pair shares one A fragment (ds_load_b64 from LDS) and
// runs two WMMA chains.  Per-lane running argmin, then shfl_xor reduction in
// the 16-lane halves, then cross-wave LDS reduction.  First-index tie-break.
// grid = N/16, block = 256.
// ---------------------------------------------------------------------------
#define ZS_LD 260   // 256 + 4 pad: conflict-free strided LDS A-fragment reads
__global__ __launch_bounds__(256) void vq_argmin(
    const float* __restrict__ Z,   // [N, 256]
    const float* __restrict__ E,   // [Kc, 256]
    const float* __restrict__ ce,  // [Kc]
    float* __restrict__ idxF,      // [N]  (float output)
    int*   __restrict__ idxI,      // [N]  (int, workspace)
    int Kc)
{
  __shared__ float zs[16 * ZS_LD];
  __shared__ float rmin[8][16];
  __shared__ int   ridx[8][16];

  const int tid  = threadIdx.x;
  const int row0 = blockIdx.x * 16;

  for (int i = tid; i < 16 * 256; i += 256)
    zs[(i >> 8) * ZS_LD + (i & 255)] = Z[(size_t)row0 * 256 + i];
  __syncthreads();

  const int lane = tid & 31, wave = tid >> 5;
  const int rsel = lane & 15, hsel = lane >> 4;

  float bmin[8];
  int   bidx[8];
#pragma unroll
  for (int v = 0; v < 8; ++v) { bmin[v] = __builtin_inff(); bidx[v] = 0; }

  const float* abase = zs + rsel * ZS_LD + 2 * hsel;

  for (int ct2 = wave; ct2 < (Kc >> 5); ct2 += 8) {
    const int col0 = ct2 * 32;
    const float* e0 = E + ((size_t)(col0 + rsel) << 8) + 2 * hsel;
    const float* e1 = e0 + (16 << 8);
    // prefetch this wave's next tile pair (speculative; tail overrun is safe)
    __builtin_prefetch(e0 + (256 << 8), 0, 1);
    __builtin_prefetch(e1 + (256 << 8), 0, 1);

    const float* ap = abase;
    v8f acc0 = {}, acc1 = {};
#pragma unroll 4
    for (int k = 0; k < 256; k += 4) {
      v2f a  = *(const v2f*)ap;       // ds_load_b64, 8B aligned, conflict-free
      v2f b0 = *(const v2f*)e0;       // E[col0+rsel][ka..ka+1]
      v2f b1 = *(const v2f*)e1;
      acc0 = __builtin_amdgcn_wmma_f32_16x16x4_f32(
          false, a, false, b0, (short)0, acc0, false, false);
      acc1 = __builtin_amdgcn_wmma_f32_16x16x4_f32(
          false, a, false, b1, (short)0, acc1, false, false);
      ap += 4; e0 += 4; e1 += 4;
    }

    const int   code0 = col0 + rsel;
    const int   code1 = code0 + 16;
    const float cen0  = ce[code0];
    const float cen1  = ce[code1];
#pragma unroll
    for (int v = 0; v < 8; ++v) {
      const float d0 = cen0 - 2.0f * acc0[v];
      if (d0 < bmin[v] || (d0 == bmin[v] && code0 < bidx[v])) {
        bmin[v] = d0; bidx[v] = code0;
      }
      const float d1 = cen1 - 2.0f * acc1[v];
      if (d1 < bmin[v] || (d1 == bmin[v] && code1 < bidx[v])) {
        bmin[v] = d1; bidx[v] = code1;
      }
    }
  }

#pragma unroll
  for (int v = 0; v < 8; ++v) {
    float m = bmin[v];
    int  mi = bidx[v];
    for (int off = 1; off < 16; off <<= 1) {   // stays inside 16-lane halves
      const float om = __shfl_xor(m, off, 32);
      const int   oi = __shfl_xor(mi, off, 32);
      if (om < m || (om == m && oi < mi)) { m = om; mi = oi; }
    }
    if (rsel == 0) { rmin[wave][v + 8 * hsel] = m; ridx[wave][v + 8 * hsel] = mi; }
  }
  __syncthreads();

  if (tid < 16) {
    float m = rmin[0][tid];
    int  mi = ridx[0][tid];
#pragma unroll
    for (int w = 1; w < 8; ++w) {
      const float om = rmin[w][tid];
      const int   oi = ridx[w][tid];
      if (om < m || (om == m && oi < mi)) { m = om; mi = oi; }
    }
    idxF[row0 + tid] = (float)mi;
    idxI[row0 + tid] = mi;
  }
}

// ---------------------------------------------------------------------------
// q = E[idx]; commit += mean((z - q)^2) partials.  grid = N, block = 256.
// ---------------------------------------------------------------------------
__global__ __launch_bounds__(256) void gather_commit(
    const float* __restrict__ Z, const float* __restrict__ E,
    const int* __restrict__ idxI, float* __restrict__ Q,
    float* __restrict__ commit, float invNC)
{
  __shared__ float red[8];
  const int tid = threadIdx.x, lane = tid & 31, w = tid >> 5;
  const size_t row = blockIdx.x;
  const int code = idxI[row];
  const float q = E[(size_t)code * 256 + tid];
  Q[row * 256 + tid] = q;
  const float d = Z[row * 256 + tid] - q;
  float s = d * d;
  for (int off = 16; off; off >>= 1) s += __shfl_xor(s, off, 32);
  if (!lane) red[w] = s;
  __syncthreads();
  if (tid == 0) {
    float t = 0.0f;
#pragma unroll
    for (int i = 0; i < 8; ++i) t += red[i];
    atomicAdd(commit, t * invNC);
  }
}

__global__ void zero_scalar(float* p) { *p = 0.0f; }

// ---------------------------------------------------------------------------
extern "C" void kernel_launch(void* const* d_in, const int* in_sizes, int n_in,
                              void* d_out, int out_size, void* d_ws, size_t ws_size,
                              hipStream_t stream) {
  (void)in_sizes; (void)n_in; (void)out_size; (void)ws_size;
  const int N = 8192, H = 512, CD = 256, Kc = 8192;
  const int DIMS[4] = {4096, 2048, 768, 1024};

  const float* E = (const float*)d_in[44];

  // Workspace: y1 [N*H] | y2 [N*CD] | ce [Kc] | idxI [N] (int)
  float* y1   = (float*)d_ws;
  float* y2   = y1 + (size_t)N * H;
  float* ce   = y2 + (size_t)N * CD;
  int*   idxI = (int*)(ce + Kc);

  codebook_norms<<<Kc, 256, 0, stream>>>(E, ce);

  size_t ooff = 0;
  for (int m = 0; m < 4; ++m) {
    const int d = DIMS[m];
    const float* x   = (const float*)d_in[m * 11 + 0];
    const float* W1  = (const float*)d_in[m * 11 + 1];
    const float* b1  = (const float*)d_in[m * 11 + 2];
    const float* g1  = (const float*)d_in[m * 11 + 3];
    const float* be1 = (const float*)d_in[m * 11 + 4];
    const float* W2  = (const float*)d_in[m * 11 + 5];
    const float* b2  = (const float*)d_in[m * 11 + 6];
    const float* g2  = (const float*)d_in[m * 11 + 7];
    const float* be2 = (const float*)d_in[m * 11 + 8];
    const float* Wd  = (const float*)d_in[m * 11 + 9];
    const float* bd  = (const float*)d_in[m * 11 + 10];

    float* z_out = (float*)d_out + ooff;  ooff += (size_t)N * CD;
    float* q_out = (float*)d_out + ooff;  ooff += (size_t)N * CD;
    float* i_out = (float*)d_out + ooff;  ooff += (size_t)N;
    float* c_out = (float*)d_out + ooff;  ooff += 1;
    float* r_out = (float*)d_out + ooff;  ooff += (size_t)N * d;

    // h1 = x @ W1 + b1 ; h = relu(LN(h1))   (LN in-place in y1)
    gemm_f32_wmma<<<dim3(H / 64, N / 64), 256, 0, stream>>>(x, W1, b1, y1, d, H);
    layernorm_rows<<<N, 256, 0, stream>>>(y1, g1, be1, y1, H, 1);

    // z = LN(h @ W2 + b2)  -> straight into d_out
    gemm_f32_wmma<<<dim3(CD / 64, N / 64), 256, 0, stream>>>(y1, W2, b2, y2, H, CD);
    layernorm_rows<<<N, 256, 0, stream>>>(y2, g2, be2, z_out, CD, 0);

    // VQ
    vq_argmin<<<N / 16, 256, 0, stream>>>(z_out, E, ce, i_out, idxI, Kc);
    zero_scalar<<<1, 1, 0, stream>>>(c_out);
    gather_commit<<<N, 256, 0, stream>>>(z_out, E, idxI, q_out, c_out,
                                         1.0f / ((float)N * (float)CD));

    // recon = q @ Wd + bd
    gemm_f32_wmma<<<dim3(d / 64, N / 64), 256, 0, stream>>>(q_out, Wd, bd, r_out, CD, d);
  }
}